// PolarToCartesianGrid_1425929142923
// MI455X (gfx1250) — compile-verified
//
#include <hip/hip_runtime.h>
#include <hip/hip_bf16.h>

// Polar -> Cartesian voxel scatter-add for MI455X (gfx1250, wave32).
//
//   polar_frames:       [16, 1, 32, 256, 128] float32  -> 16 x 1,048,576 cells
//   flat_voxel_indices: [1,048,576] int                 (values < 8,192,000)
//   output:             [16, 1, 80, 320, 320] float32  -> 16 x 8,192,000 voxels
//
// Memory-bound (~600 MB traffic -> ~26 us floor @ 23.3 TB/s). CDNA5 levers:
//   1) zero_fill: non-temporal B128 streaming stores (don't churn 192MB L2).
//   2) scatter:   wave32 segmented scan (ds_bpermute) compacts runs of equal
//      voxel indices so only run-tail lanes issue no-return
//      global_atomic_add_f32 (STOREcnt path) -> up to 32x fewer atomics on
//      the heavily-aliased low-range cells.
//   3) per-batch intensity stream double-buffered through LDS with
//      global_load_async_to_lds_b32 + s_wait_asynccnt (ASYNCcnt pipeline,
//      zero VGPR pressure, DMA overlaps the scan ALU work).

#define N_EL     32
#define N_R      256
#define N_AZ     128
#define N_CELLS  (N_EL * N_R * N_AZ)        // 1,048,576
#define N_VOX    (320 * 320 * 80)           // 8,192,000
#define N_BATCH  16
#define BLK      256

typedef __attribute__((ext_vector_type(4))) float v4f;

#ifndef __has_builtin
#define __has_builtin(x) 0
#endif

#if __has_builtin(__builtin_amdgcn_global_load_async_to_lds_b32)
#define HAVE_ASYNC_LDS 1
#else
#define HAVE_ASYNC_LDS 0
#endif

#if __has_builtin(__builtin_amdgcn_s_wait_asynccnt)
#define WAIT_ASYNCCNT(n) __builtin_amdgcn_s_wait_asynccnt(n)
#else
#define WAIT_ASYNCCNT(n) asm volatile("s_wait_asynccnt %0" ::"i"(n) : "memory")
#endif

// Builtin wants addrspace(1) int* (global src) and addrspace(3) int* (LDS dst).
typedef __attribute__((address_space(1))) int* gint_t;
typedef __attribute__((address_space(3))) int* lint_t;

// ---------------------------------------------------------------------------
// Kernel 1: stream zeros into the 524 MB output with non-temporal B128 stores.
// ---------------------------------------------------------------------------
__global__ void zero_fill_kernel(v4f* __restrict__ out, int n4) {
    int i      = blockIdx.x * blockDim.x + threadIdx.x;
    int stride = gridDim.x * blockDim.x;
    v4f z = (v4f)(0.0f);
    for (; i < n4; i += stride) {
        __builtin_nontemporal_store(z, out + i);   // global_store_b128 th:TH_STORE_NT
    }
}

// ---------------------------------------------------------------------------
// Kernel 2: scatter-add with wave32 run compaction + async-LDS batch pipeline.
// ---------------------------------------------------------------------------
__global__ void scatter_add_kernel(const float* __restrict__ polar,
                                   const int*   __restrict__ vidx,
                                   float*       __restrict__ out) {
    __shared__ float stage[2][BLK];                           // double buffer, 2 KB

    const int tid  = threadIdx.x;
    const int c    = blockIdx.x * BLK + tid;                  // cell id (grid exact)
    const int lane = tid & 31;                                // wave32 lane

    const int idx = vidx[c];

    // Run detection: index stream is azimuth-contiguous -> equal-index runs.
    // Structure is batch-invariant, computed once.
    const int  prevIdx = __shfl_up(idx, 1);
    const int  nextIdx = __shfl_down(idx, 1);
    const bool head    = (lane == 0)  || (prevIdx != idx);
    const bool tail    = (lane == 31) || (nextIdx != idx);

    const unsigned int headBits = (unsigned int)__ballot(head);
    const unsigned int below    = headBits & (0xFFFFFFFFu >> (31 - lane));
    const int runHead           = 31 - __clz(below);          // lane 0 is always a head

#if HAVE_ASYNC_LDS
    // Prologue: async-stage batch 0 (each lane stages exactly its own word;
    // ASYNCcnt is per-wave and async loads retire in order, so no barrier).
    __builtin_amdgcn_global_load_async_to_lds_b32(
        (gint_t)(polar + c), (lint_t)&stage[0][tid], 0, 0);
#endif

    #pragma unroll
    for (int b = 0; b < N_BATCH; ++b) {
#if HAVE_ASYNC_LDS
        if (b + 1 < N_BATCH) {
            __builtin_amdgcn_global_load_async_to_lds_b32(
                (gint_t)(polar + (b + 1) * N_CELLS + c),
                (lint_t)&stage[(b + 1) & 1][tid], 0, 0);
        }
        if (b + 1 < N_BATCH) { WAIT_ASYNCCNT(1); }  // oldest (batch b) landed
        else                 { WAIT_ASYNCCNT(0); }
        float v = stage[b & 1][tid];                // ds_load_b32
#else
        float v = polar[b * N_CELLS + c];
        if (b + 1 < N_BATCH) {
            __builtin_prefetch(&polar[(b + 1) * N_CELLS + c], 0, 1); // global_prefetch_b8
        }
#endif

        // Segmented inclusive scan (sum) over my run [runHead .. lane].
        float s = v;
        #pragma unroll
        for (int off = 1; off < 32; off <<= 1) {
            float o = __shfl_up(s, off);
            if (lane - off >= runHead) s += o;
        }

        if (tail) {
            // Result unused -> no-return global_atomic_add_f32 (STOREcnt path).
            atomicAdd(&out[b * N_VOX + idx], s);
        }
    }
}

// ---------------------------------------------------------------------------
extern "C" void kernel_launch(void* const* d_in, const int* in_sizes, int n_in,
                              void* d_out, int out_size, void* d_ws, size_t ws_size,
                              hipStream_t stream) {
    const float* polar = (const float*)d_in[0];   // [16,1,32,256,128] f32
    const int*   vidx  = (const int*)d_in[1];     // [1,048,576] int
    float*       out   = (float*)d_out;           // [16,1,80,320,320] f32

    // 1) Zero the output (harness poisons it; scatter touches <=1M of 8.19M voxels).
    const int n4 = out_size >> 2;                 // 32,768,000 float4s
    zero_fill_kernel<<<dim3(8192), dim3(BLK), 0, stream>>>((v4f*)out, n4);

    // 2) Scatter-add all cells, all batches.
    scatter_add_kernel<<<dim3(N_CELLS / BLK), dim3(BLK), 0, stream>>>(polar, vidx, out);
}